// Varifold_loss_45904610460381
// MI455X (gfx1250) — compile-verified
//
#include <hip/hip_runtime.h>
#include <hip/hip_bf16.h>

typedef float v2f __attribute__((ext_vector_type(2)));
typedef float v8f __attribute__((ext_vector_type(8)));

#define NV   5023
#define NF   9976
#define NFP  9984               // padded to multiple of 16
#define NT   (NFP / 16)         // 624 tiles per dimension
#define TGRP (NT / 8)           // 78 row-tile groups (8 waves / block)

// log2(e) folded constants:
//   E  = exp(-156.25 * D2)          -> exp2(KC2*D2), KC2 = -156.25*log2e
//   Kn = exp(4*(U-1))               -> exp2(KN*U + KN0)
#define KC2  (-225.42110013890053f)   // -156.25 * 1.4426950408889634
#define C2S  (450.84220027780106f)    // -2*KC2 (coefficient on x.y inside D2)
#define KN   (5.7707801635558536f)
#define KN0  (-5.7707801635558536f)

__device__ __forceinline__ float fast_exp2(float x) {
#if defined(__HIP_DEVICE_COMPILE__) && __has_builtin(__builtin_amdgcn_exp2f)
    return __builtin_amdgcn_exp2f(x);
#else
    return exp2f(x);
#endif
}

// d_ws layout: 16 meshes x 8 fields x NFP floats, then NFP zeros.
// fields: 0=CX 1=CY 2=CZ 3=NX 4=NY 5=NZ 6=A(=KC2*|C|^2) 7=L(area)
#define ZOFS ((size_t)16 * 8 * NFP)

__global__ void vf_zero(float* p, int n) {
    int i = blockIdx.x * blockDim.x + threadIdx.x;
    if (i < n) p[i] = 0.0f;
}

__global__ __launch_bounds__(256) void vf_features(const float* __restrict__ pred,
                                                   const float* __restrict__ targ,
                                                   const int*   __restrict__ faces,
                                                   float* __restrict__ W) {
    int gid = blockIdx.x * blockDim.x + threadIdx.x;     // 16*NFP items
    int m = gid / NFP;
    int t = gid - m * NFP;
    if (m >= 16) return;

    float cx = 0.f, cy = 0.f, cz = 0.f;
    float nx = 0.f, ny = 0.f, nz = 0.f;
    float av = 0.f, lv = 0.f;

    if (t < NF) {
        const float* V = (m < 8) ? pred + (size_t)m * NV * 3
                                 : targ + (size_t)(m - 8) * NV * 3;
        int f0 = faces[3 * t + 0], f1 = faces[3 * t + 1], f2 = faces[3 * t + 2];
        float x0 = V[3 * f0 + 0], y0 = V[3 * f0 + 1], z0 = V[3 * f0 + 2];
        float x1 = V[3 * f1 + 0], y1 = V[3 * f1 + 1], z1 = V[3 * f1 + 2];
        float x2 = V[3 * f2 + 0], y2 = V[3 * f2 + 1], z2 = V[3 * f2 + 2];

        cx = (x0 + x1 + x2) * (1.0f / 3.0f);
        cy = (y0 + y1 + y2) * (1.0f / 3.0f);
        cz = (z0 + z1 + z2) * (1.0f / 3.0f);

        float e1x = x1 - x0, e1y = y1 - y0, e1z = z1 - z0;
        float e2x = x2 - x0, e2y = y2 - y0, e2z = z2 - z0;
        float rx = 0.5f * (e1y * e2z - e1z * e2y);
        float ry = 0.5f * (e1z * e2x - e1x * e2z);
        float rz = 0.5f * (e1x * e2y - e1y * e2x);
        lv = sqrtf(rx * rx + ry * ry + rz * rz);
        float inv = 1.0f / lv;
        nx = rx * inv; ny = ry * inv; nz = rz * inv;
        av = KC2 * (cx * cx + cy * cy + cz * cz);
    }

    float* B = W + (size_t)m * 8 * NFP;
    B[0 * NFP + t] = cx;  B[1 * NFP + t] = cy;  B[2 * NFP + t] = cz;
    B[3 * NFP + t] = nx;  B[4 * NFP + t] = ny;  B[5 * NFP + t] = nz;
    B[6 * NFP + t] = av;  B[7 * NFP + t] = lv;
}

// One wave per 16-row strip; loops over all 624 column tiles.
// grid.x = 8 frames * 3 pairs * TGRP ; block = 256 (8 waves).
__global__ __launch_bounds__(256) void vf_main(const float* __restrict__ W,
                                               float* __restrict__ out) {
    int b     = blockIdx.x;
    int frame = b / (3 * TGRP);
    int rem   = b - frame * (3 * TGRP);
    int pair  = rem / TGRP;            // 0: P.P  1: T.T  2: P.T
    int tg    = rem - pair * TGRP;

    int tid  = threadIdx.x;
    int wave = tid >> 5;
    int lane = tid & 31;
    int half = lane >> 4;              // K-half for A/B layout
    int ln   = lane & 15;

    int rowbase = (tg * 8 + wave) * 16;

    int mx = (pair == 1) ? frame + 8 : frame;       // row mesh
    int my = (pair == 0) ? frame : frame + 8;       // col mesh
    float scale = ((pair == 2) ? -2.0f : 1.0f) * 0.125f;   // mean over 8 frames

    const float* R  = W + (size_t)mx * 8 * NFP;     // row-side fields
    const float* Q  = W + (size_t)my * 8 * NFP;     // col-side fields
    const float* ZZ = W + ZOFS;                     // NFP zeros

    const float* RCX = R + 0 * NFP; const float* RCY = R + 1 * NFP; const float* RCZ = R + 2 * NFP;
    const float* RNX = R + 3 * NFP; const float* RNY = R + 4 * NFP; const float* RNZ = R + 5 * NFP;
    const float* RA  = R + 6 * NFP; const float* RL  = R + 7 * NFP;
    const float* QA  = Q + 6 * NFP; const float* QL  = Q + 7 * NFP;

    // Loop-invariant column-side bases for the per-half K layout:
    // lanes 0-15 load (x, y); lanes 16-31 load (z, 0).
    const float* PBx = half ? (Q + 2 * NFP) : (Q + 0 * NFP);
    const float* PBy = half ? ZZ            : (Q + 1 * NFP);
    const float* PNx = half ? (Q + 5 * NFP) : (Q + 3 * NFP);
    const float* PNy = half ? ZZ            : (Q + 4 * NFP);

    // A (16x4 f32): lanes 0-15 hold row M=ln, K=0,1 ; lanes 16-31 hold K=2,3
    // Pre-scaled so the WMMA output is already the exp2 argument:
    //   centers A *= C2S  -> argC = C2S*(x.y) + (a_i + a_j)   (C operand)
    //   normals A *= KN   -> argN = KN *(u.v) + KN0           (C operand)
    int i = rowbase + ln;
    v2f Ac, An;
    Ac.x = (half ? RCZ[i] : RCX[i]) * C2S;
    Ac.y = (half ? 0.0f   : RCY[i]) * C2S;
    An.x = (half ? RNZ[i] : RNX[i]) * KN;
    An.y = (half ? 0.0f   : RNY[i]) * KN;

    float a_row[8], l_row[8];
#pragma unroll
    for (int r = 0; r < 8; ++r) {
        int ir = rowbase + r + 8 * half;   // C/D layout: M = r (+8 for upper lane half)
        a_row[r] = RA[ir];
        l_row[r] = RL[ir];
    }

    float acc[8] = {0.f, 0.f, 0.f, 0.f, 0.f, 0.f, 0.f, 0.f};
    const v8f cn = {KN0, KN0, KN0, KN0, KN0, KN0, KN0, KN0};

#pragma unroll 2
    for (int jt = 0; jt < NT; ++jt) {
        int j = jt * 16 + ln;
        v2f Bc, Bn;
        Bc.x = PBx[j];  Bc.y = PBy[j];
        Bn.x = PNx[j];  Bn.y = PNy[j];
        float a_col = QA[j];
        float l_col = QL[j];

        v8f baseC;
#pragma unroll
        for (int r = 0; r < 8; ++r) baseC[r] = a_row[r] + a_col;

        // argC[r] = C2S*(C_i.C_j) + a_i + a_j ; argN[r] = KN*(N_i.N_j) + KN0
        v8f argC = __builtin_amdgcn_wmma_f32_16x16x4_f32(false, Ac, false, Bc,
                                                         (short)0, baseC, false, false);
        v8f argN = __builtin_amdgcn_wmma_f32_16x16x4_f32(false, An, false, Bn,
                                                         (short)0, cn, false, false);

#pragma unroll
        for (int r = 0; r < 8; ++r) {
            float E  = fast_exp2(argC[r]);
            float Kn = fast_exp2(argN[r]);
            float E2 = E * E, E4 = E2 * E2, E8 = E4 * E4, E16 = E8 * E8;
            float ks = __builtin_fmaf(E4, 0.25f, E);
            ks = __builtin_fmaf(E16, 0.0625f, ks);
            acc[r] = __builtin_fmaf(ks * Kn, l_col, acc[r]);
        }
    }

    float tot = 0.f;
#pragma unroll
    for (int r = 0; r < 8; ++r) tot = __builtin_fmaf(acc[r], l_row[r], tot);

    // wave32 reduction
#pragma unroll
    for (int off = 16; off > 0; off >>= 1)
        tot += __shfl_xor(tot, off, 32);

    if (lane == 0) atomicAdd(out, tot * scale);
}

extern "C" void kernel_launch(void* const* d_in, const int* in_sizes, int n_in,
                              void* d_out, int out_size, void* d_ws, size_t ws_size,
                              hipStream_t stream) {
    const float* pred  = (const float*)d_in[0];
    const float* targ  = (const float*)d_in[1];
    const int*   faces = (const int*)d_in[2];
    float* out = (float*)d_out;
    float* W   = (float*)d_ws;   // needs (16*8+1)*NFP*4 = ~5.15 MB

    vf_zero<<<1, 64, 0, stream>>>(out, out_size);
    vf_zero<<<(NFP + 255) / 256, 256, 0, stream>>>(W + ZOFS, NFP);
    vf_features<<<(16 * NFP) / 256, 256, 0, stream>>>(pred, targ, faces, W);
    vf_main<<<8 * 3 * TGRP, 256, 0, stream>>>(W, out);
}